// DirectWarper_57363583205742
// MI455X (gfx1250) — compile-verified
//
#include <hip/hip_runtime.h>

#define B_ 8
#define H_ 384
#define W_ 1280
#define C_ 3
#define N_ (H_ * W_)            // 491520
#define WBLK (W_ / 256)         // 5 blocks per image row, 256 threads each

typedef __attribute__((ext_vector_type(2))) float v2f;
typedef __attribute__((ext_vector_type(8))) float v8f;

// ---------------------------------------------------------------------------
// Pass 0: per-batch compose  rot = K @ R @ K^-1  (3x3),  tr = K @ t  (3x1)
// params layout per batch (16 floats): rot[9], tr[3], pad[4]
// ---------------------------------------------------------------------------
__global__ void dw_pose(const float* __restrict__ pose,
                        const float* __restrict__ Km,
                        const float* __restrict__ Ki,
                        float* __restrict__ params) {
    int b = blockIdx.x;
    if (threadIdx.x != 0) return;
    const float* P  = pose + b * 12;   // (3,4) row-major
    const float* K  = Km   + b * 9;
    const float* Kv = Ki   + b * 9;
    float M1[9];                       // R @ Kinv
    for (int r = 0; r < 3; ++r)
        for (int c = 0; c < 3; ++c) {
            float s = 0.f;
            for (int k = 0; k < 3; ++k) s += P[r * 4 + k] * Kv[k * 3 + c];
            M1[r * 3 + c] = s;
        }
    float* o = params + b * 16;
    for (int r = 0; r < 3; ++r)
        for (int c = 0; c < 3; ++c) {
            float s = 0.f;
            for (int k = 0; k < 3; ++k) s += K[r * 3 + k] * M1[k * 3 + c];
            o[r * 3 + c] = s;
        }
    for (int r = 0; r < 3; ++r) {
        float s = 0.f;
        for (int k = 0; k < 3; ++k) s += K[r * 3 + k] * P[k * 4 + 3];
        o[9 + r] = s;
    }
}

// ---------------------------------------------------------------------------
// Pass 1: init 64-bit z|index keys to all-ones ("empty"), 128-bit stores
// ---------------------------------------------------------------------------
__global__ void dw_init(ulonglong2* __restrict__ keys, int npairs) {
    int i = blockIdx.x * blockDim.x + threadIdx.x;
    if (i < npairs)
        keys[i] = ulonglong2{0xFFFFFFFFFFFFFFFFull, 0xFFFFFFFFFFFFFFFFull};
}

// ---------------------------------------------------------------------------
// Pass 2: WMMA point transform + 64-bit atomic-min splat.
// key = (float_bits(z) << 32) | point_index  ->  min key == (min z, min idx),
// bit-exact with reference z-buffer + argmin tie-break (z > eps > 0 always).
// Grid: (W/256, H, B) -> no integer divisions; wave-uniform batch; full EXEC.
// ---------------------------------------------------------------------------
__global__ void dw_scatter(const float* __restrict__ depth,
                           const float* __restrict__ params,
                           unsigned long long* __restrict__ keys) {
    const int j   = blockIdx.x * 256 + threadIdx.x;     // column
    const int i   = blockIdx.y;                         // row
    const int b   = blockIdx.z;                         // batch
    const int p   = i * W_ + j;                         // point index in batch
    const float d = depth[b * N_ + p];
    const int lane = threadIdx.x & 31;

    // ---- A matrix (16x4 f32, documented layout):
    //   lanes 0-15 : v0 = A[m][0], v1 = A[m][1]
    //   lanes 16-31: v0 = A[m][2], v1 = A[m][3]
    // Rows 0..2 = [rot | tr]; rows 3..15 zero.
    const float* prm = params + b * 16;                 // same b across a wave
    int  m  = lane & 15;
    bool hi = lane >= 16;
    float a0 = 0.f, a1 = 0.f;
    if (m < 3) {
        a0 = hi ? prm[m * 3 + 2] : prm[m * 3 + 0];
        a1 = hi ? prm[9 + m]     : prm[m * 3 + 1];
    }
    v2f A = {a0, a1};

    // ---- B matrices (4x16 f32): rows (j*d, i*d, d, 1) for 16 points each.
    //   v0: lanes 0-15 = K0 row, lanes 16-31 = K2 row
    //   v1: lanes 0-15 = K1 row, lanes 16-31 = K3 row
    float jd  = (float)j * d;
    float id_ = (float)i * d;
    float xjd = __shfl(jd,  (lane + 16) & 31, 32);      // partner half-wave
    float xid = __shfl(id_, (lane + 16) & 31, 32);
    float xd  = __shfl(d,   (lane + 16) & 31, 32);

    v2f B1 = { hi ? xd : jd,  hi ? 1.0f : id_ };        // points 0-15
    v2f B2 = { hi ? d  : xjd, hi ? 1.0f : xid };        // points 16-31
    v8f Cz = {};

    // D rows 0,1,2 = x,y,z of point N=lane (lanes 0-15 of D-VGPRs 0..2)
    v8f D1 = __builtin_amdgcn_wmma_f32_16x16x4_f32(
        false, A, false, B1, (short)0, Cz, false, false);
    v8f D2 = __builtin_amdgcn_wmma_f32_16x16x4_f32(
        false, A, false, B2, (short)0, Cz, false, false);

    float x2 = __shfl(D2[0], (lane + 16) & 31, 32);     // move pts 16-31 up
    float y2 = __shfl(D2[1], (lane + 16) & 31, 32);
    float z2 = __shfl(D2[2], (lane + 16) & 31, 32);
    float x = hi ? x2 : D1[0];
    float y = hi ? y2 : D1[1];
    float z = hi ? z2 : D1[2];

    const float eps = 1e-3f;
    float zs = fmaxf(z, eps);
    int u = __float2int_rn(x / zs);                     // round-half-even,
    int v = __float2int_rn(y / zs);                     // IEEE-exact divide
    bool valid = (z > eps) & (u >= 0) & (u < W_) & (v >= 0) & (v < H_);
    if (valid) {
        unsigned long long key =
            ((unsigned long long)__float_as_uint(z) << 32) | (unsigned int)p;
        atomicMin(&keys[b * N_ + v * W_ + u], key);     // L2-resident RMW
    }
}

// ---------------------------------------------------------------------------
// Pass 3: per-pixel gather of winner color + depth. Grid: (W/256, H, B).
// ---------------------------------------------------------------------------
__global__ void dw_gather(const unsigned long long* __restrict__ keys,
                          const float* __restrict__ img,
                          float* __restrict__ out_img,
                          float* __restrict__ out_depth) {
    const int j = blockIdx.x * 256 + threadIdx.x;
    const int i = blockIdx.y;
    const int b = blockIdx.z;
    const int p = i * W_ + j;
    const int gid = b * N_ + p;

    unsigned long long key = keys[gid];
    unsigned int zbits = (unsigned int)(key >> 32);
    bool hit = zbits != 0xFFFFFFFFu;
    unsigned int idx = (unsigned int)key;
    const float* imb = img + (size_t)b * C_ * N_;
    float c0 = 0.f, c1 = 0.f, c2 = 0.f, dz = 0.f;
    if (hit) {
        c0 = imb[idx];
        c1 = imb[N_ + idx];
        c2 = imb[2 * N_ + idx];
        dz = __uint_as_float(zbits);
    }
    size_t ob = (size_t)b * C_ * N_;
    out_img[ob + p]          = c0;
    out_img[ob + N_ + p]     = c1;
    out_img[ob + 2 * N_ + p] = c2;
    out_depth[gid]           = dz;
}

// ---------------------------------------------------------------------------
extern "C" void kernel_launch(void* const* d_in, const int* in_sizes, int n_in,
                              void* d_out, int out_size, void* d_ws, size_t ws_size,
                              hipStream_t stream) {
    const float* depth = (const float*)d_in[0];   // (B,H,W)
    const float* img   = (const float*)d_in[1];   // (B,C,H,W)
    const float* pose  = (const float*)d_in[2];   // (B,3,4)
    const float* Km    = (const float*)d_in[3];   // (B,3,3)
    const float* Ki    = (const float*)d_in[4];   // (B,3,3)

    unsigned long long* keys = (unsigned long long*)d_ws;          // B*N u64
    float* params = (float*)((char*)d_ws + (size_t)B_ * N_ * 8);   // B*16 f32

    float* out_img   = (float*)d_out;                              // B*C*N
    float* out_depth = out_img + (size_t)B_ * C_ * N_;             // B*N

    const int total  = B_ * N_;                                    // 3,932,160
    const int npairs = total / 2;                                  // 128b fills
    dim3 grid(WBLK, H_, B_);

    dw_pose   <<<B_,           32,  0, stream>>>(pose, Km, Ki, params);
    dw_init   <<<npairs / 256, 256, 0, stream>>>((ulonglong2*)keys, npairs);
    dw_scatter<<<grid,         256, 0, stream>>>(depth, params, keys);
    dw_gather <<<grid,         256, 0, stream>>>(keys, img, out_img, out_depth);
}